// R2NPropositionalLayer_90202903151286
// MI455X (gfx1250) — compile-verified
//
#include <hip/hip_runtime.h>
#include <stdint.h>

// R2N propositional layer: gather 3x64 columns per row, elementwise fuzzy-logic.
// Pure bandwidth problem (~0.45 GB/launch, ~0.4 flop/byte -> ~19us at 23.3 TB/s).
// CDNA5 path: global_load_async_to_lds_b128 (ASYNCcnt) in GVS mode (SGPR base +
// 32-bit VGPR offset) with th:TH_LOAD_NT stages row tiles into LDS with
// exactly-once, 16B/lane coalesced global reads; gathers are then ds_loads.
// Outputs use non-temporal stores (streamed, never re-read; working set > L2).
// WMMA is intentionally not used: there is no reduction dimension.

#define C_DIM   256
#define T_DIM   64
#define ROWS    32                    // rows staged per block
#define THREADS 256
#define X_BYTES (ROWS * C_DIM * 4)    // 32 KB of x tile in LDS
#define B_BYTES (ROWS * C_DIM)        // 8 KB of bool tile in LDS

__global__ __launch_bounds__(THREADS)
void r2n_prop_kernel(const float* __restrict__ x,
                     const unsigned char* __restrict__ cb,
                     const int* __restrict__ i_idx,
                     const int* __restrict__ j_idx,
                     const int* __restrict__ k_idx,
                     float* __restrict__ out,
                     int Brows)
{
    __shared__ __align__(16) unsigned char smem[X_BYTES + B_BYTES];

    const int tid  = threadIdx.x;
    const int row0 = blockIdx.x * ROWS;

    // Low 32 bits of the flat shared-aperture address == workgroup-relative
    // LDS byte offset on gfx1250 (aperture lives in addr[63:32]).
    const uint32_t lds_base = (uint32_t)(uintptr_t)(&smem[0]);
    const uint32_t o16      = (uint32_t)tid * 16u;

    // ---- async stage (GVS mode): 32 KB of x + 8 KB of bools into LDS ----
    // SGPR pair holds the block-uniform tile base; one 32-bit VGPR carries
    // the lane offset. With lds_base==0 the LDS address equals the global
    // offset for the x region, so both "v" operands can share a register.
    {
        const unsigned long long xg =
            (unsigned long long)(uintptr_t)x + (unsigned long long)row0 * (C_DIM * 4);
        #pragma unroll
        for (int c = 0; c < X_BYTES / (THREADS * 16); ++c) {   // 8 issues
            const uint32_t off = (uint32_t)c * 4096u + o16;    // 16B/lane
            const uint32_t la  = lds_base + off;
            asm volatile("global_load_async_to_lds_b128 %0, %1, %2 th:TH_LOAD_NT"
                         :: "v"(la), "v"(off), "s"(xg) : "memory");
        }
        const unsigned long long bg =
            (unsigned long long)(uintptr_t)cb + (unsigned long long)row0 * C_DIM;
        #pragma unroll
        for (int c = 0; c < B_BYTES / (THREADS * 16); ++c) {   // 2 issues
            const uint32_t off = (uint32_t)c * 4096u + o16;
            const uint32_t la  = lds_base + X_BYTES + off;
            asm volatile("global_load_async_to_lds_b128 %0, %1, %2 th:TH_LOAD_NT"
                         :: "v"(la), "v"(off), "s"(bg) : "memory");
        }
    }

    // Overlap index loads (tiny, reused by all 8192 blocks -> keep cacheable).
    const int t  = tid & (T_DIM - 1);   // output column 0..63
    const int rg = tid >> 6;            // row group 0..3
    const int ii = i_idx[t];
    const int jj = j_idx[t];
    const int kk = k_idx[t];

    asm volatile("s_wait_asynccnt 0x0" ::: "memory");
    __syncthreads();

    const float*         lx = (const float*)smem;
    const unsigned char* lb = smem + X_BYTES;
    const size_t bbBase = (size_t)Brows * T_DIM;

    #pragma unroll
    for (int u = 0; u < ROWS / 4; ++u) {
        const int r      = u * 4 + rg;
        const int rowOff = r * C_DIM;

        const float xi = lx[rowOff + ii];
        const float xj = lx[rowOff + jj];
        const float xk = lx[rowOff + kk];
        const int   bi = lb[rowOff + ii];
        const int   bj = lb[rowOff + jj];
        const int   bk = lb[rowOff + kk];

        // emb = 1 - (1 - xi*xj) * (1 - (1 - xk)) == 1 - (1 - xi*xj)*xk
        const float emb = 1.0f - (1.0f - xi * xj) * xk;
        const float bbv = (((bi != 0) && (bj != 0)) || (bk == 0)) ? 1.0f : 0.0f;

        const size_t o = (size_t)(row0 + r) * T_DIM + (size_t)t;
        __builtin_nontemporal_store(emb, &out[o]);           // first output (B,T)
        __builtin_nontemporal_store(bbv, &out[bbBase + o]);  // second output
    }
}

extern "C" void kernel_launch(void* const* d_in, const int* in_sizes, int n_in,
                              void* d_out, int out_size, void* d_ws, size_t ws_size,
                              hipStream_t stream)
{
    const float*         x  = (const float*)d_in[0];
    const unsigned char* cb = (const unsigned char*)d_in[1]; // jnp bool: 1 byte/elem
    const int* ii = (const int*)d_in[2];
    const int* jj = (const int*)d_in[3];
    const int* kk = (const int*)d_in[4];
    float* out = (float*)d_out;

    const int Brows = in_sizes[0] / C_DIM;   // 262144
    const int tiles = Brows / ROWS;          // 8192 blocks
    hipLaunchKernelGGL(r2n_prop_kernel, dim3(tiles), dim3(THREADS), 0, stream,
                       x, cb, ii, jj, kk, out, Brows);
}